// sinkhorn_81905026334859
// MI455X (gfx1250) — compile-verified
//
#include <hip/hip_runtime.h>

// Sinkhorn-Knopp on MI455X (gfx1250, wave32).
// d_in[0]: C [32,2048,2048] f32; d_in[1]: epsilon (1 f32)
// d_out:   u [32*2048] then v [32*2048] (f32)
// d_ws:    column-sum accumulator c [32*2048] f32 (256 KB)

typedef __attribute__((ext_vector_type(2))) float v2f;
typedef __attribute__((ext_vector_type(8))) float v8f;

#define BATCH 32
#define NROW 2048
#define MCOL 2048
#define ROWS_PER_WG 64
#define WG 256            // 8 wave32 waves
#define WAVES 8
#define ROWS_PER_WAVE 8   // 64 / 8
#define COLS_PER_WAVE 256 // 2048 / 8
#define N_ITER 10

__global__ __launch_bounds__(WG) void sinkhorn_init(float* __restrict__ u,
                                                    float* __restrict__ v,
                                                    float* __restrict__ c) {
    int i = blockIdx.x * blockDim.x + threadIdx.x;   // covers BATCH*2048
    u[i] = 1.0f;
    v[i] = 1.0f;
    c[i] = 0.0f;
}

// One Sinkhorn half-step pair over a 64-row block of one batch:
//   phase 1: r_i = sum_j C[i][j] v[j] ; u_new_i = u_i / (r_i + eps)
//   phase 2: c[j] += sum_{i in block} u_new_i * C[i][j]   (WMMA f32 16x16x4)
__global__ __launch_bounds__(WG) void sinkhorn_iter(const float* __restrict__ C,
                                                    const float* __restrict__ eps_p,
                                                    float* __restrict__ u,
                                                    const float* __restrict__ v,
                                                    float* __restrict__ c) {
    __shared__ float4 v_s4[MCOL / 4];     // 8 KB: v staged once per WG
    __shared__ float  u_s[ROWS_PER_WG];   // u_new for this row block

    const int b    = blockIdx.y;
    const int i0   = blockIdx.x * ROWS_PER_WG;
    const int t    = threadIdx.x;
    const int lane = t & 31;
    const int wave = t >> 5;
    const float eps = *eps_p;

    const float*  Cb = C + (size_t)b * NROW * MCOL;
    const float4* v4 = reinterpret_cast<const float4*>(v + (size_t)b * MCOL);

    for (int k = t; k < MCOL / 4; k += WG) v_s4[k] = v4[k];
    __syncthreads();

    // ---- phase 1: coalesced HBM stream, 512B per wave per load ----
    for (int rr = 0; rr < ROWS_PER_WAVE; ++rr) {
        const int row = wave * ROWS_PER_WAVE + rr;              // block-local 0..63
        const float4* Crow4 =
            reinterpret_cast<const float4*>(Cb + (size_t)(i0 + row) * MCOL);
        float4 acc = make_float4(0.f, 0.f, 0.f, 0.f);
        #pragma unroll
        for (int jt = 0; jt < (MCOL / 4) / 32; ++jt) {
            __builtin_prefetch(Crow4 + jt * 32 + lane + 256, 0, 0); // speculative stream-ahead
            float4 x = Crow4[jt * 32 + lane];
            float4 w = v_s4[jt * 32 + lane];
            acc.x += x.x * w.x; acc.y += x.y * w.y;
            acc.z += x.z * w.z; acc.w += x.w * w.w;
        }
        float s = acc.x + acc.y + acc.z + acc.w;
        #pragma unroll
        for (int off = 16; off > 0; off >>= 1) s += __shfl_xor(s, off, 32);
        if (lane == 0) {
            const int gi = b * NROW + i0 + row;
            const float un = u[gi] / (s + eps);
            u[gi]  = un;
            u_s[row] = un;
        }
    }
    __syncthreads();

    // ---- phase 2: u^T x Cblock via V_WMMA_F32_16X16X4_F32 (reads hit L2) ----
    // A[m][k] = u_s[ib+k] (broadcast over m): lanes 0-15 hold K=0,1; lanes 16-31 K=2,3.
    // B[k][n] = C[i0+ib+k][j0+n]: row striped across lanes, lane-half offsets K by 2.
    // D[m][n] = c_partial[j0+n], replicated over m.
    const int n  = lane & 15;
    const int kh = lane >> 4;   // 0 or 1
    for (int g = 0; g < COLS_PER_WAVE / 16; ++g) {
        const int j0 = wave * COLS_PER_WAVE + g * 16;
        v8f d = {};
        #pragma unroll
        for (int ib = 0; ib < ROWS_PER_WG; ib += 4) {
            const int r0 = ib + 2 * kh;
            v2f a, bb;
            a[0] = u_s[r0];
            a[1] = u_s[r0 + 1];
            const float* p = Cb + (size_t)(i0 + r0) * MCOL + j0 + n;
            bb[0] = p[0];
            bb[1] = p[MCOL];
            d = __builtin_amdgcn_wmma_f32_16x16x4_f32(
                    /*neg_a=*/false, a, /*neg_b=*/false, bb,
                    /*c_mod=*/(short)0, d, /*reuse_a=*/false, /*reuse_b=*/false);
        }
        if (lane < 16) atomicAdd(&c[(size_t)b * MCOL + j0 + lane], d[0]);
    }
}

__global__ __launch_bounds__(WG) void sinkhorn_vupd(const float* __restrict__ eps_p,
                                                    float* __restrict__ v,
                                                    float* __restrict__ c) {
    int i = blockIdx.x * blockDim.x + threadIdx.x;   // covers BATCH*MCOL
    const float eps = *eps_p;
    v[i] = v[i] / (c[i] + eps);
    c[i] = 0.0f;   // re-arm accumulator for next iteration
}

extern "C" void kernel_launch(void* const* d_in, const int* in_sizes, int n_in,
                              void* d_out, int out_size, void* d_ws, size_t ws_size,
                              hipStream_t stream) {
    (void)in_sizes; (void)n_in; (void)out_size; (void)ws_size;
    const float* C   = (const float*)d_in[0];
    const float* eps = (const float*)d_in[1];
    float* u = (float*)d_out;                       // [32*2048]
    float* v = (float*)d_out + BATCH * NROW;        // [32*2048]
    float* c = (float*)d_ws;                        // [32*2048]

    sinkhorn_init<<<(BATCH * NROW) / WG, WG, 0, stream>>>(u, v, c);
    for (int it = 0; it < N_ITER; ++it) {
        sinkhorn_iter<<<dim3(NROW / ROWS_PER_WG, BATCH), WG, 0, stream>>>(C, eps, u, v, c);
        sinkhorn_vupd<<<(BATCH * MCOL) / WG, WG, 0, stream>>>(eps, v, c);
    }
}